// BERTSpanNER_2061584302550
// MI455X (gfx1250) — compile-verified
//
#include <hip/hip_runtime.h>
#include <math.h>

// ---- problem constants (from reference) ----
#define BB   4
#define SS   1024
#define HH   400
#define NLBL 16
#define NTAG 65          // 1 + 4*16
#define NEGV (-1.0e9f)
#define EPSV (1.0e-8f)

typedef float v2f __attribute__((ext_vector_type(2)));
typedef float v4f __attribute__((ext_vector_type(4)));
typedef float v8f __attribute__((ext_vector_type(8)));

// =====================================================================
// Kernel 1: tag GEMM via V_WMMA_F32_16X16X4_F32 + fused log_softmax and
// per-label reductions (begin/end/inside).
//   grid: 64 blocks x 128 threads (4 waves/block, 1 wave = one 16-row tile)
// =====================================================================
__global__ __launch_bounds__(128)
void ner_gemm_softmax_kernel(const float* __restrict__ x,
                             const float* __restrict__ W,
                             const float* __restrict__ bias,
                             float* __restrict__ g_begin,
                             float* __restrict__ g_end,
                             float* __restrict__ g_inside) {
    __shared__ float lds[4][16][82];           // 82-f padded rows: conflict-free

    const int lane = threadIdx.x & 31;
    const int wave = threadIdx.x >> 5;                 // 0..3
    const int tile = blockIdx.x * 4 + wave;            // 0..255
    const int row0 = tile * 16;                        // global flat row (b*S+s)
    const int m    = lane & 15;
    const int kk   = (lane >> 4) << 1;                 // 0 for lanes 0-15, 2 for 16-31

    const float* xrow = x + (size_t)(row0 + m) * HH;

    v8f acc[5] = {};                                   // 5 N-tiles cover 80 >= 65 cols

    // K loop: 100 steps of K=4.  A lane loads a contiguous float2 of x.
    for (int k0 = 0; k0 < HH; k0 += 4) {
        const v2f a = *(const v2f*)(xrow + k0 + kk);
#pragma unroll
        for (int nt = 0; nt < 5; ++nt) {
            const int n = nt * 16 + m;
            v2f bv;
            if (n < NTAG) {
                bv.x = W[(size_t)(k0 + kk)     * NTAG + n];
                bv.y = W[(size_t)(k0 + kk + 1) * NTAG + n];
            } else {
                bv.x = 0.f; bv.y = 0.f;
            }
            acc[nt] = __builtin_amdgcn_wmma_f32_16x16x4_f32(
                false, a, false, bv, (short)0, acc[nt], false, false);
        }
    }

    // Spill accumulators to LDS: vgpr r, lanes 0-15 -> M=r, lanes 16-31 -> M=r+8
#pragma unroll
    for (int nt = 0; nt < 5; ++nt) {
#pragma unroll
        for (int r = 0; r < 8; ++r) {
            const int row = (lane < 16) ? r : (r + 8);
            const int n   = nt * 16 + m;
            if (n < 80) lds[wave][row][n] = acc[nt][r];
        }
    }
    __syncthreads();

    // One lane per row: softmax over 65 tags + per-label reductions.
    if (lane < 16) {
        const int r    = lane;
        const int grow = row0 + r;                     // b*S + s
        float* Lrow = &lds[wave][r][0];

        float mx = -INFINITY;
        for (int n = 0; n < NTAG; ++n) {
            const float v = Lrow[n] + bias[n];
            Lrow[n] = v;
            mx = fmaxf(mx, v);
        }
        float sum = 0.f;
        for (int n = 0; n < NTAG; ++n) sum += expf(Lrow[n] - mx);
        const float lse = mx + logf(sum);

        for (int l = 0; l < NLBL; ++l) {
            const float It = Lrow[1 + 4 * l] - lse;
            const float Bt = Lrow[2 + 4 * l] - lse;
            const float Lt = Lrow[3 + 4 * l] - lse;
            const float Ut = Lrow[4 + 4 * l] - lse;

            // logaddexp(B,U), logaddexp(L,U)
            float mb = fmaxf(Bt, Ut);
            float be = mb + logf(expf(Bt - mb) + expf(Ut - mb));
            float me = fmaxf(Lt, Ut);
            float en = me + logf(expf(Lt - me) + expf(Ut - me));
            // logsumexp(I,B,L,U)
            float mi = fmaxf(fmaxf(It, Bt), fmaxf(Lt, Ut));
            float ins = mi + logf(expf(It - mi) + expf(Bt - mi) +
                                  expf(Lt - mi) + expf(Ut - mi));

            g_begin [(size_t)grow * NLBL + l] = be;
            g_end   [(size_t)grow * NLBL + l] = en;
            g_inside[(size_t)grow * NLBL + l] = ins;
        }
    }
}

// =====================================================================
// Kernel 2: exclusive-prefix cumsum over seq for each (b,l).
//   64 blocks x 32 threads; one wave = one scan of length 1024 via
//   wave32 shuffle scans (32 chunks of 32).
// =====================================================================
__global__ __launch_bounds__(32)
void ner_cumsum_kernel(const float* __restrict__ inside,
                       float* __restrict__ cum) {
    const int b    = blockIdx.x >> 4;
    const int l    = blockIdx.x & 15;
    const int lane = threadIdx.x;

    if (lane == 0) cum[(size_t)(b * (SS + 1)) * NLBL + l] = 0.f;

    float carry = 0.f;
    for (int t = 0; t < SS / 32; ++t) {
        const int s = t * 32 + lane;
        float v = inside[(size_t)(b * SS + s) * NLBL + l];
#pragma unroll
        for (int d = 1; d < 32; d <<= 1) {
            const float u = __shfl_up(v, d, 32);
            if (lane >= d) v += u;
        }
        const float out = carry + v;
        cum[(size_t)(b * (SS + 1) + s + 1) * NLBL + l] = out;
        carry = __shfl(out, 31, 32);
    }
}

// =====================================================================
// Kernel 3: span scoring — the 256 MiB streaming-store kernel.
//   4096 blocks (one per (b,i)) x 256 threads; each thread owns one
//   l-quad (float4) for 64 j's per iteration; nontemporal b128 stores.
// =====================================================================
__global__ __launch_bounds__(256)
void ner_span_kernel(const float* __restrict__ g_begin,
                     const float* __restrict__ g_end,
                     const float* __restrict__ cum,
                     float* __restrict__ out) {
    const unsigned bi = blockIdx.x;            // 0..4095 == b*S + i
    const unsigned b  = bi >> 10;
    const unsigned i  = bi & 1023u;
    const int t    = threadIdx.x;
    const int lq   = t & 3;                    // which float4 of the 16 labels
    const int joff = t >> 2;                   // 0..63

    const v4f bI = *(const v4f*)(g_begin + (size_t)bi * NLBL + lq * 4);
    const v4f cI = *(const v4f*)(cum + (size_t)(b * (SS + 1) + i) * NLBL + lq * 4);

    v4f neg4; neg4.x = NEGV; neg4.y = NEGV; neg4.z = NEGV; neg4.w = NEGV;

    v4f* orow = (v4f*)(out + (size_t)bi * SS * NLBL);

    for (int j0 = 0; j0 < SS; j0 += 64) {
        const int j = j0 + joff;
        v4f res;
        if (j < (int)i) {
            res = neg4;
        } else {
            const v4f cJ = *(const v4f*)(cum   + (size_t)(b * (SS + 1) + j + 1) * NLBL + lq * 4);
            const v4f eJ = *(const v4f*)(g_end + (size_t)(b * SS + j) * NLBL + lq * 4);
            res.x = fminf(fminf(fminf(cJ.x - cI.x, -EPSV), bI.x), eJ.x);
            res.y = fminf(fminf(fminf(cJ.y - cI.y, -EPSV), bI.y), eJ.y);
            res.z = fminf(fminf(fminf(cJ.z - cI.z, -EPSV), bI.z), eJ.z);
            res.w = fminf(fminf(fminf(cJ.w - cI.w, -EPSV), bI.w), eJ.w);
        }
        __builtin_nontemporal_store(res, &orow[(size_t)j * 4 + lq]);
    }
}

// =====================================================================
extern "C" void kernel_launch(void* const* d_in, const int* in_sizes, int n_in,
                              void* d_out, int out_size, void* d_ws, size_t ws_size,
                              hipStream_t stream) {
    const float* x    = (const float*)d_in[0];
    // d_in[1] = mask: unused by the reference computation.
    const float* W    = (const float*)d_in[2];
    const float* bias = (const float*)d_in[3];
    float* out = (float*)d_out;

    // workspace layout (floats), all 16B aligned
    float* ws       = (float*)d_ws;
    float* g_inside = ws;                                  // 4096*16
    float* g_begin  = g_inside + (size_t)BB * SS * NLBL;   // 4096*16
    float* g_end    = g_begin  + (size_t)BB * SS * NLBL;   // 4096*16
    float* g_cum    = g_end    + (size_t)BB * SS * NLBL;   // 4*1025*16

    // Phase 1: WMMA GEMM + softmax reductions. 256 row-tiles, 4 per block.
    ner_gemm_softmax_kernel<<<64, 128, 0, stream>>>(x, W, bias,
                                                    g_begin, g_end, g_inside);
    // Phase 2: 64 wave-scans.
    ner_cumsum_kernel<<<BB * NLBL, 32, 0, stream>>>(g_inside, g_cum);
    // Phase 3: 256 MiB streaming output, one block per (b,i).
    ner_span_kernel<<<BB * SS, 256, 0, stream>>>(g_begin, g_end, g_cum, out);
}